// REINFORCE_63694365000237
// MI455X (gfx1250) — compile-verified
//
#include <hip/hip_runtime.h>
#include <math.h>

// Problem constants (match reference)
#define B_ 16
#define S_ 4096
#define V_ 128
#define TRUNC_ 11
#define T_ (S_ - TRUNC_)          // 4085
#define C_ 128                    // time chunks (pass-1 parallelism: B_*C_ = 2048 waves)
#define CH_ (S_ / C_)             // 32 timesteps per chunk
#define X_ (B_ * V_)              // 2048 output streams (b,v)

typedef __attribute__((ext_vector_type(2))) float v2f;
typedef __attribute__((ext_vector_type(8))) float v8f;

// logf(0.99)
#define LOG_GAMMA (-0.010050335853501441f)

__device__ __forceinline__ float log_sigmoid_f(float x) {
    // log_sigmoid(x) = min(x,0) - log1p(exp(-|x|)), numerically stable
    return fminf(x, 0.0f) - log1pf(__expf(-fabsf(x)));
}

// ---------------------------------------------------------------------------
// Pass 1: one wave per (b, chunk). Each lane owns 4 consecutive v (float4
// loads -> one wave covers all V=128 per timestep, fully coalesced 512B/instr).
// Sequential scan over the 32 timesteps of the chunk:
//   p    += lp' (masked prefix of log_probs)
//   wls   = gamma^(S-u) * log_sigmoid(logit)
//   acc  += wls * p ;  swl += wls
// Emits per-chunk triples to workspace for the WMMA combine pass.
// ---------------------------------------------------------------------------
__global__ void __launch_bounds__(256)
reinforce_pass1(const float* __restrict__ log_probs,   // [S,B,V]
                const float* __restrict__ logits,      // [B,S,V]
                float* __restrict__ accA,              // [C][X]
                float* __restrict__ lpsA,              // [C][X]
                float* __restrict__ swlA)              // [C][X]
{
    const int wave = (blockIdx.x * blockDim.x + threadIdx.x) >> 5;
    const int lane = threadIdx.x & 31;
    const int b    = wave & (B_ - 1);
    const int c    = wave >> 4;
    const int v0   = lane * 4;
    const int t0   = c * CH_;

    const float* lgBase = logits    + (size_t)b * S_ * V_ + v0;   // stride V_ per step
    const float* lpBase = log_probs + (size_t)b * V_       + v0;  // stride B_*V_ per step

    float4 p   = {0.f, 0.f, 0.f, 0.f};
    float4 acc = {0.f, 0.f, 0.f, 0.f};
    float4 swl = {0.f, 0.f, 0.f, 0.f};

    for (int u = t0; u < t0 + CH_; ++u) {
        const float4 lg = *(const float4*)(lgBase + (size_t)u * V_);
        const float4 lp = *(const float4*)(lpBase + (size_t)u * (B_ * V_));

        // Stream-ahead prefetch (global_prefetch_b8); clamp to stay in-bounds.
        int up = u + 8; if (up > S_ - 1) up = S_ - 1;
        __builtin_prefetch(lgBase + (size_t)up * V_, 0, 0);
        __builtin_prefetch(lpBase + (size_t)up * (B_ * V_), 0, 0);

        const float msk = (u < T_) ? 1.0f : 0.0f;     // lp'[u]
        p.x += lp.x * msk; p.y += lp.y * msk;
        p.z += lp.z * msk; p.w += lp.w * msk;

        const float wu = __expf((float)(S_ - u) * LOG_GAMMA);  // gamma^(S-u)
        float4 wl;
        wl.x = wu * log_sigmoid_f(lg.x);
        wl.y = wu * log_sigmoid_f(lg.y);
        wl.z = wu * log_sigmoid_f(lg.z);
        wl.w = wu * log_sigmoid_f(lg.w);

        acc.x = fmaf(wl.x, p.x, acc.x); acc.y = fmaf(wl.y, p.y, acc.y);
        acc.z = fmaf(wl.z, p.z, acc.z); acc.w = fmaf(wl.w, p.w, acc.w);
        swl.x += wl.x; swl.y += wl.y; swl.z += wl.z; swl.w += wl.w;
    }

    const size_t o = (size_t)c * X_ + (size_t)b * V_ + v0;
    *(float4*)(accA + o) = acc;
    *(float4*)(lpsA + o) = p;      // chunk sum of masked log_probs
    *(float4*)(swlA + o) = swl;    // chunk sum of wls
}

// ---------------------------------------------------------------------------
// Pass 2: inter-chunk exclusive scan + combine, via fp32 WMMA.
// Per wave: 16 streams (n = lane&15), 128 chunks in 8 blocks of 16.
// For each block:  D(16x16) = Lstrict(16x16) @ lps_block(16x16) + carry
// realized as 4x V_WMMA_F32_16X16X4_F32 (K split 16 = 4x4), carry broadcast
// through the C operand.  D[m,n] = exclusive prefix of chunk sums.
// Then e += acc + swl * D (elementwise, reduced over m per lane + one shuffle).
// Carry for next block taken from row m=15: D[15] + lps[15].
// ---------------------------------------------------------------------------
__global__ void __launch_bounds__(256)
reinforce_pass2(const float* __restrict__ accA,
                const float* __restrict__ lpsA,
                const float* __restrict__ swlA,
                float* __restrict__ out)               // [X_] = [B,V]
{
    const int wave = (blockIdx.x * blockDim.x + threadIdx.x) >> 5;
    const int lane = threadIdx.x & 31;
    const int half = lane >> 4;        // C/D rows: 0 -> M=0..7, 1 -> M=8..15
    const int nl   = lane & 15;        // column index n (stream within tile)
    const int x    = wave * 16 + nl;   // global stream id for this lane's column
    const int m    = nl;               // A-matrix row held by this lane

    float carry = 0.0f;                // Sum of chunk lp-sums over previous blocks
    float e     = 0.0f;

    for (int blk = 0; blk < C_ / 16; ++blk) {
        const int cb = blk * 16;

        v8f d;
        #pragma unroll
        for (int i = 0; i < 8; ++i) d[i] = carry;   // C[m,n] = carry[n]

        #pragma unroll
        for (int g = 0; g < 4; ++g) {
            // K indices held by this lane for both A and B operands:
            const int k0 = g * 4 + half * 2;
            const int k1 = k0 + 1;
            v2f a, bm;
            a.x = (k0 < m) ? 1.0f : 0.0f;           // strict lower-triangular ones
            a.y = (k1 < m) ? 1.0f : 0.0f;
            bm.x = lpsA[(size_t)(cb + k0) * X_ + x];
            bm.y = lpsA[(size_t)(cb + k1) * X_ + x];
            // D = A*B + D   (8-arg form: neg_a, A, neg_b, B, c_mod, C, reuse_a, reuse_b)
            d = __builtin_amdgcn_wmma_f32_16x16x4_f32(
                    false, a, false, bm, (short)0, d, false, false);
        }

        float top = 0.0f;
        #pragma unroll
        for (int i = 0; i < 8; ++i) {
            const int mm = half * 8 + i;            // global row in tile
            const size_t idx = (size_t)(cb + mm) * X_ + x;
            e = fmaf(swlA[idx], d[i], e) + accA[idx];
            if (mm == 15) top = d[i] + lpsA[idx];   // inclusive total of this block
        }
        // broadcast block total (held by upper-half lanes) to whole wave
        carry = __shfl(top, nl + 16, 32);
    }

    // Combine the two row-halves (m=0..7 in lanes 0-15, m=8..15 in lanes 16-31)
    const float eo = __shfl(e, nl + 16, 32);
    if (lane < 16) out[x] = e + eo;
}

// ---------------------------------------------------------------------------
extern "C" void kernel_launch(void* const* d_in, const int* in_sizes, int n_in,
                              void* d_out, int out_size, void* d_ws, size_t ws_size,
                              hipStream_t stream) {
    const float* log_probs = (const float*)d_in[0];  // [S,B,V]
    const float* logits    = (const float*)d_in[1];  // [B,S,V]

    float* wsf  = (float*)d_ws;                      // needs 3*C_*X_*4 = 3 MB
    float* accA = wsf;
    float* lpsA = wsf + (size_t)C_ * X_;
    float* swlA = wsf + (size_t)2 * C_ * X_;

    // Pass 1: B_*C_ = 2048 waves -> 65536 threads -> 256 blocks of 256
    reinforce_pass1<<<(B_ * C_ * 32) / 256, 256, 0, stream>>>(
        log_probs, logits, accA, lpsA, swlA);

    // Pass 2: X_/16 = 128 waves -> 4096 threads -> 16 blocks of 256
    reinforce_pass2<<<(X_ / 16 * 32) / 256, 256, 0, stream>>>(
        accA, lpsA, swlA, (float*)d_out);
}